// Advection_70892730187952
// MI455X (gfx1250) — compile-verified
//
#include <hip/hip_runtime.h>

// Advection stencil: out[b,i,j] = (s[i+1,j]-s[i,j])*vy + (s[i,j+1]-s[i,j])*vx
// Shapes: state [32,512,512,1] f32, velocity [32,512,512,2] f32, out [32,512,512,1] f32.
// Memory-bound: ~134 MB min traffic @ 23.3 TB/s => ~5.8 us floor. Strategy:
//  - stage state rows into LDS with gfx1250 async-to-LDS b128 copies (single global fetch,
//    row reuse served from LDS), s_wait_asynccnt + barrier
//  - velocity / output are pure streams -> non-temporal b128 accesses
//  - everything coalesced 16B per lane

#define HH 512
#define WW 512
#define TM 8          // output rows per block
#define THREADS 256   // 8 wave32 waves

typedef float v4f __attribute__((ext_vector_type(4)));

__global__ __launch_bounds__(THREADS) void advect_kernel(
    const float* __restrict__ state,   // [B,H,W]
    const float* __restrict__ vel,     // [B,H,W,2]
    float* __restrict__ out)           // [B,H,W]
{
    __shared__ float tile[(TM + 1) * WW];   // 9 rows x 512 f32 = 18 KB

    const int tid = threadIdx.x;
    const int i0  = blockIdx.x * TM;        // first output row of this tile
    const int b   = blockIdx.y;

    // ---------------- Phase 1: async-copy state rows i0..i0+TM into LDS ----------------
    const unsigned long long sbase =
        (unsigned long long)(const void*)state + (unsigned long long)((size_t)b * HH * WW * 4u);

    const int totalVec4 = (TM + 1) * (WW / 4);   // 1152 b128 transfers
    #pragma unroll
    for (int k = 0; k < (totalVec4 + THREADS - 1) / THREADS; ++k) {
        int idx = tid + k * THREADS;
        if (idx < totalVec4) {
            int row  = idx >> 7;              // /128 (W/4)
            int c4   = (idx & 127) << 2;      // column (floats)
            int grow = i0 + row;
            if (grow > HH - 1) grow = HH - 1; // symmetric pad: clamp halo row
            unsigned long long gaddr = sbase + (unsigned long long)(grow * WW + c4) * 4u;
            // Flat pointers into LDS carry the LDS byte offset in their low 32 bits.
            unsigned lds_off = (unsigned)(size_t)(const void*)&tile[row * WW + c4];
            asm volatile("global_load_async_to_lds_b128 %0, %1, off"
                         :: "v"(lds_off), "v"(gaddr)
                         : "memory");
        }
    }
    asm volatile("s_wait_asynccnt 0" ::: "memory");
    __syncthreads();

    // ---------------- Phase 2: compute 8 rows, 4 columns per thread per step ----------------
    const float* __restrict__ vbase = vel + (size_t)b * HH * WW * 2;
    float*       __restrict__ obase = out + (size_t)b * HH * WW;

    const int j0     = (tid & 127) << 2;   // column of this thread's 4-vector
    const int rlocal = tid >> 7;           // 0 or 1: two rows processed per step

    #pragma unroll
    for (int step = 0; step < TM / 2; ++step) {
        const int r = step * 2 + rlocal;   // local row in tile, 0..7
        const int i = i0 + r;              // global row

        // state from LDS (ds_load_b128 + halo scalar)
        v4f s0 = *(const v4f*)&tile[r * WW + j0];
        v4f s1 = *(const v4f*)&tile[(r + 1) * WW + j0];
        int je = j0 + 4;
        if (je > WW - 1) je = WW - 1;      // symmetric pad: dx = 0 at last column
        float sE = tile[r * WW + je];

        // velocity: interleaved (vy,vx) pairs; 2 x b128 non-temporal (no reuse)
        const v4f* vp = (const v4f*)(vbase + ((size_t)i * WW + j0) * 2);
        v4f va = __builtin_nontemporal_load(vp);       // vy0 vx0 vy1 vx1
        v4f vb = __builtin_nontemporal_load(vp + 1);   // vy2 vx2 vy3 vx3

        float dy0 = s1.x - s0.x, dy1 = s1.y - s0.y, dy2 = s1.z - s0.z, dy3 = s1.w - s0.w;
        float dx0 = s0.y - s0.x, dx1 = s0.z - s0.y, dx2 = s0.w - s0.z, dx3 = sE - s0.w;

        v4f o;
        o.x = fmaf(dy0, va.x, dx0 * va.y);
        o.y = fmaf(dy1, va.z, dx1 * va.w);
        o.z = fmaf(dy2, vb.x, dx2 * vb.y);
        o.w = fmaf(dy3, vb.z, dx3 * vb.w);

        __builtin_nontemporal_store(o, (v4f*)&obase[(size_t)i * WW + j0]);
    }
}

extern "C" void kernel_launch(void* const* d_in, const int* in_sizes, int n_in,
                              void* d_out, int out_size, void* d_ws, size_t ws_size,
                              hipStream_t stream) {
    const float* state = (const float*)d_in[0];   // [B,512,512,1] f32
    const float* vel   = (const float*)d_in[1];   // [B,512,512,2] f32
    float* out         = (float*)d_out;           // [B,512,512,1] f32

    const int B = in_sizes[0] / (HH * WW);        // 32

    dim3 grid(HH / TM, B);                        // (64, 32)
    dim3 block(THREADS);
    advect_kernel<<<grid, block, 0, stream>>>(state, vel, out);
}